// Conv2d_20426864460058
// MI455X (gfx1250) — compile-verified
//
#include <hip/hip_runtime.h>

typedef __attribute__((ext_vector_type(2))) float v2f;
typedef __attribute__((ext_vector_type(8))) float v8f;

#define HW (128 * 128)

// One block = one (batch, 16x16 spatial tile). 8 waves of 32.
// Stage A: Z[halo_pixel][c_mid] = relu(bn2( (W1 @ relu(bn1(x)))  ))  (18x18 halo, zero-pad -> relu(bn2(0)))
// Stage B: out = W2 @ shift(Z) + b2   (shift == per-channel LDS gather offset)
__global__ __launch_bounds__(256, 1)
void shiftnet_fused_kernel(const float* __restrict__ x,
                           const float* __restrict__ gamma1, const float* __restrict__ beta1,
                           const float* __restrict__ mean1,  const float* __restrict__ var1,
                           const float* __restrict__ w1,
                           const float* __restrict__ gamma2, const float* __restrict__ beta2,
                           const float* __restrict__ mean2,  const float* __restrict__ var2,
                           const float* __restrict__ w2,     const float* __restrict__ b2,
                           float* __restrict__ out)
{
    constexpr int PSTRIDE  = 68;   // floats per halo pixel in LDS (68 % 64 == 4 -> no bank conflicts)
    constexpr int NPIX     = 324;  // 18*18 halo pixels
    constexpr int NPIX_PAD = 336;  // 21 n-tiles of 16

    extern __shared__ float lds[];
    float* Z      = lds;                      // NPIX_PAD * PSTRIDE floats
    float* s_inv1 = Z + NPIX_PAD * PSTRIDE;   // 64
    float* s_sh1  = s_inv1 + 64;              // 64
    float* s_inv2 = s_sh1 + 64;               // 64
    float* s_sh2  = s_inv2 + 64;              // 64

    const int tid  = threadIdx.x;
    const int lane = tid & 31;
    const int wv   = tid >> 5;
    const int half = lane >> 4;    // which half-wave (K offset +2 in f32 WMMA frags)
    const int ln   = lane & 15;    // M (A-frag) / N (B,C,D-frag) index
    const int mt   = wv & 3;       // m-tile: output channels mt*16..mt*16+15
    const int ws   = wv >> 2;      // n-tile phase (0/1), two waves per m-tile

    const int wt = blockIdx.x * 16;
    const int ht = blockIdx.y * 16;
    const int b  = blockIdx.z;

    // Fused BN affine tables (eval-mode): y = x*inv + sh
    if (tid < 64) {
        float i1 = gamma1[tid] * rsqrtf(var1[tid] + 1e-5f);
        s_inv1[tid] = i1;
        s_sh1[tid]  = beta1[tid] - mean1[tid] * i1;
        float i2 = gamma2[tid] * rsqrtf(var2[tid] + 1e-5f);
        s_inv2[tid] = i2;
        s_sh2[tid]  = beta2[tid] - mean2[tid] * i2;
    }
    __syncthreads();

    const int m  = mt * 16 + ln;        // A-fragment row = output channel
    const int cb = mt * 16 + 8 * half;  // C/D-fragment channel base (VGPR i <-> ch cb+i)

    // ---------------- Stage A ----------------
    // Preload W1 A-fragments + bn1 constants for this lane's K channels (k = 4*ks + 2*half + v)
    v2f a1[16], i1v[16], s1v[16];
    #pragma unroll
    for (int ks = 0; ks < 16; ++ks) {
        const float* pw = w1 + m * 64 + ks * 4 + 2 * half;
        a1[ks].x = pw[0]; a1[ks].y = pw[1];
        int c = ks * 4 + 2 * half;
        i1v[ks].x = s_inv1[c];  i1v[ks].y = s_inv1[c + 1];
        s1v[ks].x = s_sh1[c];   s1v[ks].y = s_sh1[c + 1];
    }
    float i2r[8], s2r[8];
    #pragma unroll
    for (int i = 0; i < 8; ++i) { i2r[i] = s_inv2[cb + i]; s2r[i] = s_sh2[cb + i]; }

    const float* xb = x + (size_t)b * 64 * HW;

    for (int t = ws; t < 21; t += 2) {             // wave-uniform loop (EXEC stays full)
        int p  = t * 16 + ln;                      // flattened halo pixel
        int hy = p / 18;
        int hx = p - hy * 18;
        int gh = ht - 1 + hy;
        int gw = wt - 1 + hx;
        bool inimg = (p < NPIX) & (gh >= 0) & (gh < 128) & (gw >= 0) & (gw < 128);
        const float* xp = xb + (inimg ? (gh * 128 + gw) : 0);   // clamped, always valid

        v8f acc = {0.f, 0.f, 0.f, 0.f, 0.f, 0.f, 0.f, 0.f};
        #pragma unroll
        for (int ks = 0; ks < 16; ++ks) {
            int c0 = (ks * 4 + 2 * half) * HW;
            float x0 = xp[c0];
            float x1 = xp[c0 + HW];
            v2f bf;  // B[k][n] = relu(bn1(x))   (garbage columns for !inimg are overwritten below)
            bf.x = fmaxf(fmaf(inimg ? x0 : 0.f, i1v[ks].x, s1v[ks].x), 0.f);
            bf.y = fmaxf(fmaf(inimg ? x1 : 0.f, i1v[ks].y, s1v[ks].y), 0.f);
            acc = __builtin_amdgcn_wmma_f32_16x16x4_f32(false, a1[ks], false, bf,
                                                        (short)0, acc, false, false);
        }
        // Apply bn2+relu; out-of-image halo pixels hold relu(bn2(0)) = relu(sh2)
        float* zp = Z + p * PSTRIDE + cb;
        #pragma unroll
        for (int i = 0; i < 8; ++i)
            zp[i] = fmaxf(inimg ? fmaf(acc[i], i2r[i], s2r[i]) : s2r[i], 0.f);
    }
    __syncthreads();

    // ---------------- Stage B ----------------
    v2f a2[16];
    #pragma unroll
    for (int ks = 0; ks < 16; ++ks) {
        const float* pw = w2 + m * 64 + ks * 4 + 2 * half;
        a2[ks].x = pw[0]; a2[ks].y = pw[1];
    }
    float br[8];
    #pragma unroll
    for (int i = 0; i < 8; ++i) br[i] = b2[cb + i];

    // Shift gather offsets per K-step: ch c -> group g=c/7 -> (ky,kx)=(g/3,g%3); ch 63 -> (0,0)
    int shof[16][2];
    #pragma unroll
    for (int ks = 0; ks < 16; ++ks) {
        #pragma unroll
        for (int v = 0; v < 2; ++v) {
            int c  = ks * 4 + 2 * half + v;
            int g  = c / 7;
            int ky = (c >= 63) ? 0 : g / 3;
            int kx = (c >= 63) ? 0 : g % 3;
            shof[ks][v] = (ky * 18 + kx) * PSTRIDE + c;
        }
    }

    for (int t = ws; t < 16; t += 2) {             // t = output row in tile, n = output col
        const float* zb = Z + (t * 18 + ln) * PSTRIDE;
        v8f acc = {0.f, 0.f, 0.f, 0.f, 0.f, 0.f, 0.f, 0.f};
        #pragma unroll
        for (int ks = 0; ks < 16; ++ks) {
            v2f bf;
            bf.x = zb[shof[ks][0]];
            bf.y = zb[shof[ks][1]];
            acc = __builtin_amdgcn_wmma_f32_16x16x4_f32(false, a2[ks], false, bf,
                                                        (short)0, acc, false, false);
        }
        float* op = out + ((size_t)b * 64 + cb) * HW + (ht + t) * 128 + (wt + ln);
        #pragma unroll
        for (int i = 0; i < 8; ++i)
            op[i * HW] = acc[i] + br[i];
    }
}

extern "C" void kernel_launch(void* const* d_in, const int* in_sizes, int n_in,
                              void* d_out, int out_size, void* d_ws, size_t ws_size,
                              hipStream_t stream) {
    const float* x      = (const float*)d_in[0];
    const float* gamma1 = (const float*)d_in[1];
    const float* beta1  = (const float*)d_in[2];
    const float* mean1  = (const float*)d_in[3];
    const float* var1   = (const float*)d_in[4];
    const float* w1     = (const float*)d_in[5];
    const float* gamma2 = (const float*)d_in[6];
    const float* beta2  = (const float*)d_in[7];
    const float* mean2  = (const float*)d_in[8];
    const float* var2   = (const float*)d_in[9];
    const float* w2     = (const float*)d_in[10];
    const float* b2     = (const float*)d_in[11];
    float* out = (float*)d_out;

    dim3 grid(128 / 16, 128 / 16, 32);                 // (w-tiles, h-tiles, batch)
    size_t shmem = (size_t)(336 * 68 + 4 * 64) * sizeof(float);  // 92,416 B
    shiftnet_fused_kernel<<<grid, dim3(256), shmem, stream>>>(
        x, gamma1, beta1, mean1, var1, w1,
        gamma2, beta2, mean2, var2, w2, b2, out);
}